// DependencyLabelClassifier_16681652977791
// MI455X (gfx1250) — compile-verified
//
#include <hip/hip_runtime.h>

typedef __attribute__((ext_vector_type(2))) float v2f;
typedef __attribute__((ext_vector_type(4))) float v4f;
typedef __attribute__((ext_vector_type(8))) float v8f;

#define BB  32
#define LL  128
#define DD  384
#define NL  64
#define NC  128   // concatenated output cols: [head(64) | dep(64)]

// ---------------------------------------------------------------------------
// Kernel 0: build Wt[d, n] (384 x 128, row-major) from W[64, 768]:
//   n <  64 : Wt[d,n] = W[n, d]        (head half)
//   n >= 64 : Wt[d,n] = W[n-64, 384+d] (dep half)
// ---------------------------------------------------------------------------
__global__ void prep_w_kernel(const float* __restrict__ W, float* __restrict__ Wt) {
    int idx = blockIdx.x * blockDim.x + threadIdx.x;
    if (idx >= DD * NC) return;
    int n = idx & (NC - 1);
    int d = idx >> 7;
    float v = (n < NL) ? W[n * (2 * DD) + d]
                       : W[(n - NL) * (2 * DD) + DD + d];
    Wt[d * NC + n] = v;
}

// ---------------------------------------------------------------------------
// Kernel 1: HD[4096,128] = X[4096,384] @ Wt[384,128] via V_WMMA_F32_16X16X4_F32
// 128-thread blocks = 4 waves; each wave owns one 16-row M tile and all 8
// N tiles (8 x v8f accumulators), 96 K-steps of K=4.
// A frag (16x4 f32): lanes 0-15 -> K={0,1}, lanes 16-31 -> K={2,3}, row = lane&15
// B frag (4x16 f32): symmetric, col = lane&15
// D (16x16 f32): lanes 0-15 -> M=r, lanes 16-31 -> M=r+8, N = lane&15
// ---------------------------------------------------------------------------
__global__ void gemm_hd_kernel(const float* __restrict__ X,
                               const float* __restrict__ Wt,
                               float* __restrict__ HD) {
    const int lane  = threadIdx.x & 31;
    const int wave  = blockIdx.x * (blockDim.x >> 5) + (threadIdx.x >> 5);
    const int mbase = wave * 16;            // 256 waves cover M=4096
    const int lrow  = lane & 15;
    const int khalf = (lane >> 4) << 1;     // 0 for lanes 0-15, 2 for 16-31

    v8f acc[8];
#pragma unroll
    for (int t = 0; t < 8; ++t)
#pragma unroll
        for (int i = 0; i < 8; ++i) acc[t][i] = 0.0f;

    const float* xrow = X + (size_t)(mbase + lrow) * DD + khalf;

    for (int k4 = 0; k4 < DD; k4 += 4) {
        v2f a;
        a.x = xrow[k4 + 0];
        a.y = xrow[k4 + 1];
        const float* wrow = Wt + (size_t)(k4 + khalf) * NC + lrow;
#pragma unroll
        for (int t = 0; t < 8; ++t) {
            v2f b;
            b.x = wrow[t * 16];
            b.y = wrow[t * 16 + NC];
            acc[t] = __builtin_amdgcn_wmma_f32_16x16x4_f32(
                false, a, false, b, (short)0, acc[t], false, false);
        }
    }

    const int mofs = (lane >> 4) << 3;      // +8 for upper half-wave
#pragma unroll
    for (int t = 0; t < 8; ++t) {
#pragma unroll
        for (int r = 0; r < 8; ++r) {
            HD[(size_t)(mbase + mofs + r) * NC + t * 16 + lrow] = acc[t][r];
        }
    }
}

// ---------------------------------------------------------------------------
// Kernel 2 (bandwidth-bound): out[b, j*L+k, n] =
//   mask(b,j,k) ? HD[b*L+k, n] + HD[b*L+j, 64+n] : -inf
// Each thread emits one v4f of output; per-block output is 4KB contiguous.
// Non-temporal stores keep the 134MB stream from thrashing L2 (HD stays hot).
// ---------------------------------------------------------------------------
__global__ void bcast_mask_kernel(const float* __restrict__ HD,
                                  const unsigned char* __restrict__ att,
                                  float* __restrict__ out) {
    int idx  = blockIdx.x * blockDim.x + threadIdx.x;
    int n4   = (idx & 15) << 2;             // n offset: 0,4,...,60
    int pair = idx >> 4;                    // b*L*L + j*L + k
    int k = pair & (LL - 1);
    int j = (pair >> 7) & (LL - 1);
    int b = pair >> 14;

    bool m = att[b * LL + j] && att[b * LL + k] && (j != k);

    const v4f h = *reinterpret_cast<const v4f*>(HD + (size_t)(b * LL + k) * NC + n4);
    const v4f d = *reinterpret_cast<const v4f*>(HD + (size_t)(b * LL + j) * NC + NL + n4);

    const float NINF = -__builtin_inff();
    v4f o;
    o.x = m ? h.x + d.x : NINF;
    o.y = m ? h.y + d.y : NINF;
    o.z = m ? h.z + d.z : NINF;
    o.w = m ? h.w + d.w : NINF;

    __builtin_nontemporal_store(o, reinterpret_cast<v4f*>(out + (size_t)pair * NL + n4));
}

// ---------------------------------------------------------------------------
extern "C" void kernel_launch(void* const* d_in, const int* in_sizes, int n_in,
                              void* d_out, int out_size, void* d_ws, size_t ws_size,
                              hipStream_t stream) {
    const float*         emb = (const float*)d_in[0];         // [32,128,384] f32
    const unsigned char* att = (const unsigned char*)d_in[1]; // [32,128] bool
    const float*         W   = (const float*)d_in[2];         // [64,768] f32
    float*               out = (float*)d_out;                 // [32,16384,64] f32

    float* Wt = (float*)d_ws;            // 384*128 floats
    float* HD = Wt + DD * NC;            // 4096*128 floats

    prep_w_kernel<<<(DD * NC + 255) / 256, 256, 0, stream>>>(W, Wt);
    gemm_hd_kernel<<<64, 128, 0, stream>>>(emb, Wt, HD);     // 256 waves
    bcast_mask_kernel<<<(BB * LL * LL * 16) / 256, 256, 0, stream>>>(HD, att, out);
}